// MaxUnpooling2D_32366873542794
// MI455X (gfx1250) — compile-verified
//
#include <hip/hip_runtime.h>

// MaxUnpooling2D scatter-add for MI455X (gfx1250).
//
// Shapes (fixed by the reference's setup_inputs):
//   inputs : (8, 128, 128, 64) fp32   -> 8,388,608 elems, 2^20 per batch
//   indices: same shape, int32, per-batch flat index into out_h*out_w*out_c
//   output : (8, 256, 256, 64) fp32   -> 33,554,432 elems, 2^22 per batch
//
// Strategy: output (134 MB) fits in the 192 MB L2 -> do plain HW float
// atomics (global_atomic_add_f32, no-return form) which resolve in L2.
// The input/index streams are read exactly once -> non-temporal loads
// (th:TH_LOAD_NT) so the 67 MB of streaming data does not evict the
// L2-resident output working set.

typedef __attribute__((ext_vector_type(4))) float float4v;
typedef __attribute__((ext_vector_type(4))) int   int4v;

__global__ __launch_bounds__(256)
void unpool_zero_fill(float4v* __restrict__ out, int n4) {
    int i = blockIdx.x * blockDim.x + threadIdx.x;
    if (i < n4) {
        float4v z;
        z.x = 0.0f; z.y = 0.0f; z.z = 0.0f; z.w = 0.0f;
        out[i] = z;   // global_store_b128
    }
}

__global__ __launch_bounds__(256)
void unpool_scatter_add(const float4v* __restrict__ in4,
                        const int4v*  __restrict__ idx4,
                        float* __restrict__ out,
                        int n4,                 // number of float4 groups in input
                        int shift_in,           // log2(elems per input batch)  = 20
                        int shift_out)          // log2(elems per output batch) = 22
{
    int t = blockIdx.x * blockDim.x + threadIdx.x;
    if (t >= n4) return;

    // Non-temporal streaming loads (B128): read-once data, keep L2 for output.
    float4v v  = __builtin_nontemporal_load(in4 + t);
    int4v   ix = __builtin_nontemporal_load(idx4 + t);

    // 4 consecutive input elements always share a batch (batch size is 2^20,
    // group stride is 4). Element index of this group's first element:
    long long e0    = (long long)t << 2;
    int       batch = (int)(e0 >> shift_in);
    long long base  = (long long)batch << shift_out;   // batch offset in output

    // No-return HW float atomics -> global_atomic_add_f32 (L2-resident RMW).
    unsafeAtomicAdd(out + (base + (long long)ix.x), v.x);
    unsafeAtomicAdd(out + (base + (long long)ix.y), v.y);
    unsafeAtomicAdd(out + (base + (long long)ix.z), v.z);
    unsafeAtomicAdd(out + (base + (long long)ix.w), v.w);
}

extern "C" void kernel_launch(void* const* d_in, const int* in_sizes, int n_in,
                              void* d_out, int out_size, void* d_ws, size_t ws_size,
                              hipStream_t stream) {
    const float4v* in4  = (const float4v*)d_in[0];
    const int4v*   idx4 = (const int4v*)d_in[1];
    float*         out  = (float*)d_out;

    const int n_in_elems  = in_sizes[0];      // 8,388,608
    const int n_out_elems = out_size;         // 33,554,432
    const int B = 8;                          // fixed by reference setup

    const int per_batch_in  = n_in_elems  / B;   // 2^20
    const int per_batch_out = n_out_elems / B;   // 2^22
    const int shift_in  = __builtin_ctz((unsigned)per_batch_in);
    const int shift_out = __builtin_ctz((unsigned)per_batch_out);

    // ---- Pass 1: zero the output (134 MB of B128 stores) ----
    {
        const int n4 = n_out_elems / 4;           // 8,388,608 float4 stores
        const int threads = 256;
        const int blocks  = (n4 + threads - 1) / threads;
        unpool_zero_fill<<<blocks, threads, 0, stream>>>((float4v*)out, n4);
    }

    // ---- Pass 2: scatter-add with L2-resident float atomics ----
    {
        const int n4 = n_in_elems / 4;            // 2,097,152 groups
        const int threads = 256;
        const int blocks  = (n4 + threads - 1) / threads;
        unpool_scatter_add<<<blocks, threads, 0, stream>>>(
            in4, idx4, out, n4, shift_in, shift_out);
    }
}